// RankCoxHazardLoss_33595234189714
// MI455X (gfx1250) — compile-verified
//
#include <hip/hip_runtime.h>
#include <hip/hip_bf16.h>

#define EPSF   1e-7f
#define BATCH  512
#define NN     256

typedef __attribute__((ext_vector_type(2))) float v2f;
typedef __attribute__((ext_vector_type(8))) float v8f;

// Cross-lane broadcast via LDS-permute hardware (no LDS memory touched).
__device__ __forceinline__ float bcast32(float v, int srcLane) {
  return __builtin_bit_cast(
      float, __builtin_amdgcn_ds_bpermute(srcLane << 2, __builtin_bit_cast(int, v)));
}

__device__ __forceinline__ float wave_sum32(float v, int lane) {
#pragma unroll
  for (int m = 1; m < 32; m <<= 1) v += bcast32(v, lane ^ m);
  return v;
}

// D = A(16x4 f32) x ones(4x16) + C : every D column holds the row sums.
__device__ __forceinline__ v8f row_accum(v2f a, v2f ones, v8f c) {
  return __builtin_amdgcn_wmma_f32_16x16x4_f32(
      /*neg_a=*/false, a, /*neg_b=*/false, ones,
      /*c_mod=*/(short)0, c, /*reuse_a=*/false, /*reuse_b=*/false);
}

// 16x16 f32 C/D layout: lane l holds D(m = r + 8*(l>>4), n = l&15) in d[r].
// With B = ones all columns are identical.  Lane l needs row m = l&15:
//   sel = d[l&7], then fetch from lane (l&7) | (((l>>3)&1) << 4).
__device__ __forceinline__ float extract_row_scalar(v8f d, int lane) {
  const int r = lane & 7;
  float sel = d[0];
  sel = (r == 1) ? (float)d[1] : sel;
  sel = (r == 2) ? (float)d[2] : sel;
  sel = (r == 3) ? (float)d[3] : sel;
  sel = (r == 4) ? (float)d[4] : sel;
  sel = (r == 5) ? (float)d[5] : sel;
  sel = (r == 6) ? (float)d[6] : sel;
  sel = (r == 7) ? (float)d[7] : sel;
  const int src = (lane & 7) | (((lane >> 3) & 1) << 4);
  return bcast32(sel, src);
}

// Total over all 16 rows of a C/D accumulator (columns identical per half).
__device__ __forceinline__ float tile_total(v8f d, int lane) {
  float t = 0.0f;
#pragma unroll
  for (int r = 0; r < 8; ++r) t += d[r];
  return t + bcast32(t, lane ^ 16);
}

__global__ __launch_bounds__(256) void cox_rank_rows(
    const float* __restrict__ pred, const float* __restrict__ target,
    const int* __restrict__ vmask, float* __restrict__ row_cox,
    float* __restrict__ row_rank, float* __restrict__ row_valid) {
  // Packed per-j data: s_tmE[j] = {tm, E}; s_pack[j] = {tm, E, R, pred}.
  __shared__ __align__(16) float2 s_tmE[NN];
  __shared__ __align__(16) float4 s_pack[NN];
  __shared__ float red[NN];
  __shared__ float wpart[8][3];

  const int b    = blockIdx.x;
  const int t    = threadIdx.x;
  const int lane = t & 31;
  const int wave = t >> 5;

  const float* pr = pred + b * NN;
  const float* tr = target + b * NN;
  const int*   vr = vmask + b * NN;

  __builtin_prefetch(pr + t, 0, 1);   // global_prefetch_b8
  __builtin_prefetch(tr + t, 0, 1);

  // ---------------- Phase 0: stage row, row-wide reductions ----------------
  const float p   = pr[t];
  const float tg  = tr[t];
  const int   vmi = vr[t];
  const float v   = vmi ? 1.0f : 0.0f;
  const float tmv = vmi ? tg : -1.0f;

  // max(tm) -> batch_max_time
  red[t] = tmv;
  __syncthreads();
#pragma unroll
  for (int s = 128; s > 0; s >>= 1) {
    if (t < s) red[t] = fmaxf(red[t], red[t + s]);
    __syncthreads();
  }
  const float bmax = red[0];
  __syncthreads();

  // max(valid pred) -> shift M for all exps
  red[t] = vmi ? p : -1e30f;
  __syncthreads();
#pragma unroll
  for (int s = 128; s > 0; s >>= 1) {
    if (t < s) red[t] = fmaxf(red[t], red[t + s]);
    __syncthreads();
  }
  float M = red[0];
  __syncthreads();
  if (M < -1e29f) M = 0.0f;   // no valid entries: keep math finite

  // sum(valid) -> valid_batch test
  red[t] = v;
  __syncthreads();
#pragma unroll
  for (int s = 128; s > 0; s >>= 1) {
    if (t < s) red[t] += red[t + s];
    __syncthreads();
  }
  const float vsum = red[0];
  __syncthreads();

  // O(N) transcendentals per row instead of O(N^2)
  const float E = vmi ? __expf(p - M) : 0.0f;  // valid-masked; E>0 <=> valid
  const float R = __expf(M - p);               // 1/exp(p - M)
  s_tmE[t]  = make_float2(tmv, E);
  s_pack[t] = make_float4(tmv, E, R, p);
  __syncthreads();

  // ---------------- Phase 1: per-wave 16-row tiles, WMMA row reductions ----
  const v2f ones  = {1.0f, 1.0f};
  const int li    = lane & 15;
  const int kbase = (lane >> 4) << 1;   // lanes 0-15: K0,K1 ; lanes 16-31: K2,K3

  v8f accL2 = {}, accRank = {}, accPair = {};
  float l1sum = 0.0f;

  for (int ib = wave; ib < 16; ib += 8) {
    const int    i  = ib * 16 + li;
    const float4 qi = s_pack[i];
    const float  t_i = qi.x, e_i = qi.y, p_i = qi.w;
    const bool   elim  = (t_i < bmax) && (t_i > 0.0f) && (e_i > 0.0f);
    const float  felim = elim ? 1.0f : 0.0f;
    const float  fvi   = (e_i > 0.0f) ? 1.0f : 0.0f;

    // Pass A: S_i = sum over risk set of E_j (masked row sum via WMMA).
    // One ds_load_b128 per chunk: {tm0, E0, tm1, E1}.
    v8f accS = {};
#pragma unroll 4
    for (int jc = 0; jc < 64; ++jc) {
      const int    j0 = jc * 4 + kbase;
      const float4 q  = *(const float4*)&s_tmE[j0];
      v2f a;
      a.x = q.y * ((q.x >= t_i) ? 1.0f : 0.0f);
      a.y = q.w * ((q.z >= t_i) ? 1.0f : 0.0f);
      accS = row_accum(a, ones, accS);
    }
    const float S       = extract_row_scalar(accS, lane);
    const float log_den = M + __logf(S);
    // Fast hardware reciprocal (v_rcp_f32); S>0 whenever the value is used.
    const float invS    = (S > 0.0f) ? __builtin_amdgcn_rcpf(S) : 0.0f;

    // loss_part1: count each i once (lower half lanes only)
    l1sum += (elim && lane < 16) ? (log_den - p_i) : 0.0f;

    // Pass B: surviving-log sum, rank numerator, pair count
#pragma unroll 4
    for (int jc = 0; jc < 64; ++jc) {
      const int    j0 = jc * 4 + kbase;
      const float4 q0 = s_pack[j0];
      const float4 q1 = s_pack[j0 + 1];
      v2f aL2, aRk, aPr;
#pragma unroll
      for (int u = 0; u < 2; ++u) {
        const float4 q = u ? q1 : q0;
        const int    j = j0 + u;
        // q = {tm_j, E_j, R_j, p_j}; valid_j <=> E_j > 0
        const float m_surv =
            ((q.x >= t_i) && (j != i) && (q.y > 0.0f)) ? felim : 0.0f;
        const float pcl = fminf(q.y * invS, 1.0f - EPSF);
        const float lv  = -__logf(1.0f - pcl + EPSF);
        const float a2  = lv * m_surv;

        const float m_tp = ((t_i > q.x) && (q.y > 0.0f)) ? fvi : 0.0f;
        const float ap   = m_tp;
        const float ar   = (e_i * q.z) * ((p_i > q.w) ? m_tp : 0.0f);

        if (u == 0) { aL2.x = a2; aPr.x = ap; aRk.x = ar; }
        else        { aL2.y = a2; aPr.y = ap; aRk.y = ar; }
      }
      accL2   = row_accum(aL2, ones, accL2);
      accRank = row_accum(aRk, ones, accRank);
      accPair = row_accum(aPr, ones, accPair);
    }
  }

  // ---------------- Phase 2: wave + block reduction, one row result -------
  const float tL2 = tile_total(accL2, lane);
  const float tRk = tile_total(accRank, lane);
  const float tPr = tile_total(accPair, lane);
  const float l1  = wave_sum32(l1sum, lane);

  if (lane == 0) {
    wpart[wave][0] = l1 + tL2;
    wpart[wave][1] = tRk;
    wpart[wave][2] = tPr;
  }
  __syncthreads();

  if (t == 0) {
    float cox = 0.0f, rk = 0.0f, prs = 0.0f;
#pragma unroll
    for (int w = 0; w < 8; ++w) {
      cox += wpart[w][0];
      rk  += wpart[w][1];
      prs += wpart[w][2];
    }
    const float vb = (vsum >= 2.0f) ? 1.0f : 0.0f;
    row_cox[b]   = cox * vb;
    row_valid[b] = vb;
    row_rank[b]  = rk / fmaxf(prs, 1.0f);
  }
}

__global__ __launch_bounds__(256) void finalize_kernel(
    const float* __restrict__ row_cox, const float* __restrict__ row_rank,
    const float* __restrict__ row_valid, float* __restrict__ out) {
  __shared__ float rc[256], rr[256], rv[256];
  const int t = threadIdx.x;
  rc[t] = row_cox[t] + row_cox[t + 256];
  rr[t] = row_rank[t] + row_rank[t + 256];
  rv[t] = row_valid[t] + row_valid[t + 256];
  __syncthreads();
#pragma unroll
  for (int s = 128; s > 0; s >>= 1) {
    if (t < s) {
      rc[t] += rc[t + s];
      rr[t] += rr[t + s];
      rv[t] += rv[t + s];
    }
    __syncthreads();
  }
  if (t == 0) out[0] = rc[0] / fmaxf(rv[0], 1.0f) + rr[0] / (float)BATCH;
}

extern "C" void kernel_launch(void* const* d_in, const int* in_sizes, int n_in,
                              void* d_out, int out_size, void* d_ws, size_t ws_size,
                              hipStream_t stream) {
  (void)in_sizes; (void)n_in; (void)out_size; (void)ws_size;
  const float* pred   = (const float*)d_in[0];
  const float* target = (const float*)d_in[1];
  const int*   vmask  = (const int*)d_in[2];

  float* ws        = (float*)d_ws;
  float* row_cox   = ws;
  float* row_rank  = ws + BATCH;
  float* row_valid = ws + 2 * BATCH;

  cox_rank_rows<<<BATCH, 256, 0, stream>>>(pred, target, vmask,
                                           row_cox, row_rank, row_valid);
  finalize_kernel<<<1, 256, 0, stream>>>(row_cox, row_rank, row_valid,
                                         (float*)d_out);
}